// TorchInterpolationModule_89816356094519
// MI455X (gfx1250) — compile-verified
//
#include <hip/hip_runtime.h>

// Masked box-filter diffusion (3 iterations) fused into a single kernel.
// HBM-bandwidth-bound: ~402 MB minimum traffic -> ~17 us at 23.3 TB/s.
// Uses the CDNA5 async global->LDS DMA path (GLOBAL_LOAD_ASYNC_TO_LDS_B32,
// tracked by ASYNCcnt, waited with S_WAIT_ASYNCCNT) for the tile+halo load;
// all 3 diffusion iterations run entirely in LDS; one global store per pixel.

#define IMG_H   512
#define IMG_W   512
#define TILE    32
#define RAD     3
#define REG     (TILE + 2 * RAD)   // 38 loaded rows/cols
#define PITCH   40                 // padded LDS row pitch (floats)
#define INTR    (REG - 2)          // 36: interior computed each iteration
#define NTHREADS 256

// ---- CDNA5 async global->LDS load (one f32 per lane), inline asm ----------
// VGLOBAL encoding: VDST = 32-bit LDS byte address, VADDR = 64-bit global
// address, 'off' = no SADDR. Generic pointers to __shared__ carry the LDS
// byte offset in their low 32 bits (ISA 10.2 aperture mapping), so a simple
// truncating cast produces the LDS address operand.
__device__ __forceinline__ void async_load_f32(const float* gsrc, float* ldst) {
    unsigned lds_addr = (unsigned)(size_t)ldst;
    asm volatile("global_load_async_to_lds_b32 %0, %1, off"
                 :: "v"(lds_addr), "v"(gsrc)
                 : "memory");
}

__device__ __forceinline__ void wait_async0() {
    asm volatile("s_wait_asynccnt 0" ::: "memory");
}

// ---------------------------------------------------------------------------
__global__ __launch_bounds__(NTHREADS)
void masked_diffuse3_kernel(const float* __restrict__ data,
                            const float* __restrict__ mask,
                            float* __restrict__ out) {
    __shared__ float sD [REG * PITCH];   // original data tile (+halo)
    __shared__ float sM [REG * PITCH];   // original mask tile (+halo)
    __shared__ float sX1[REG * PITCH];
    __shared__ float sN1[REG * PITCH];
    __shared__ float sX2[REG * PITCH];
    __shared__ float sN2[REG * PITCH];

    const int tid   = threadIdx.x;
    const int tileX = blockIdx.x;                 // 16 tiles
    const int tileY = blockIdx.y;                 // 16 tiles
    const int img   = blockIdx.z;                 // B*C images
    const long long imgBase = (long long)img * (IMG_H * IMG_W);

    const int gy0 = tileY * TILE - RAD;
    const int gx0 = tileX * TILE - RAD;

    // ---- Phase 0: async DMA tile+halo into LDS (zeros outside the image) --
    for (int c = tid; c < REG * REG; c += NTHREADS) {
        const int ly = c / REG;
        const int lx = c - ly * REG;
        const int gy = gy0 + ly;
        const int gx = gx0 + lx;
        const int off = ly * PITCH + lx;
        if ((unsigned)gy < (unsigned)IMG_H && (unsigned)gx < (unsigned)IMG_W) {
            const long long g = imgBase + (long long)gy * IMG_W + gx;
            async_load_f32(data + g, &sD[off]);
            async_load_f32(mask + g, &sM[off]);
        } else {
            sD[off] = 0.0f;     // SAME zero padding
            sM[off] = 0.0f;
        }
    }
    wait_async0();          // this wave's DMA landed in LDS
    __syncthreads();        // all waves' tiles visible

    // ---- Iteration 1: (D*M, M) -> (X1, N1), interior rings >= 1 -----------
    for (int c = tid; c < INTR * INTR; c += NTHREADS) {
        const int iy = c / INTR + 1;
        const int ix = c - (iy - 1) * INTR + 1;
        const int off = iy * PITCH + ix;
        float sd = 0.0f, sm = 0.0f;
        #pragma unroll
        for (int dy = -1; dy <= 1; ++dy) {
            #pragma unroll
            for (int dx = -1; dx <= 1; ++dx) {
                const int o = off + dy * PITCH + dx;
                const float m = sM[o];
                sd += sD[o] * m;
                sm += m;
            }
        }
        const float cd = sd * (1.0f / 9.0f);
        const float cm = sm * (1.0f / 9.0f);
        const float xo = sD[off] * sM[off];
        sX1[off] = (cm > 0.0f) ? (cd / fmaxf(cm, 1e-8f)) : xo;
        sN1[off] = fminf(cm, 1.0f);
    }
    __syncthreads();

    // ---- Iteration 2: (X1, N1) -> (X2, N2) --------------------------------
    for (int c = tid; c < INTR * INTR; c += NTHREADS) {
        const int iy = c / INTR + 1;
        const int ix = c - (iy - 1) * INTR + 1;
        const int off = iy * PITCH + ix;
        float sd = 0.0f, sm = 0.0f;
        #pragma unroll
        for (int dy = -1; dy <= 1; ++dy) {
            #pragma unroll
            for (int dx = -1; dx <= 1; ++dx) {
                const int o = off + dy * PITCH + dx;
                sd += sX1[o];
                sm += sN1[o];
            }
        }
        const float cd = sd * (1.0f / 9.0f);
        const float cm = sm * (1.0f / 9.0f);
        sX2[off] = (cm > 0.0f) ? (cd / fmaxf(cm, 1e-8f)) : sX1[off];
        sN2[off] = fminf(cm, 1.0f);
    }
    __syncthreads();

    // ---- Iteration 3 fused with final select + single global store --------
    for (int c = tid; c < TILE * TILE; c += NTHREADS) {
        const int oy = c / TILE;
        const int ox = c - oy * TILE;
        const int off = (oy + RAD) * PITCH + (ox + RAD);
        float sd = 0.0f, sm = 0.0f;
        #pragma unroll
        for (int dy = -1; dy <= 1; ++dy) {
            #pragma unroll
            for (int dx = -1; dx <= 1; ++dx) {
                const int o = off + dy * PITCH + dx;
                sd += sX2[o];
                sm += sN2[o];
            }
        }
        const float cd = sd * (1.0f / 9.0f);
        const float cm = sm * (1.0f / 9.0f);
        const float x3 = (cm > 0.0f) ? (cd / fmaxf(cm, 1e-8f)) : sX2[off];
        const float d0 = sD[off];
        const float m0 = sM[off];
        const long long g = imgBase
                          + (long long)(tileY * TILE + oy) * IMG_W
                          + (tileX * TILE + ox);
        out[g] = (m0 > 0.0f) ? d0 : x3;
    }
}

// ---------------------------------------------------------------------------
extern "C" void kernel_launch(void* const* d_in, const int* in_sizes, int n_in,
                              void* d_out, int out_size, void* d_ws, size_t ws_size,
                              hipStream_t stream) {
    const float* data = (const float*)d_in[0];
    const float* mask = (const float*)d_in[1];
    float* out = (float*)d_out;

    const int n_img = in_sizes[0] / (IMG_H * IMG_W);   // B*C = 128
    dim3 grid(IMG_W / TILE, IMG_H / TILE, n_img);      // 16 x 16 x 128
    dim3 block(NTHREADS);
    masked_diffuse3_kernel<<<grid, block, 0, stream>>>(data, mask, out);
}